// GNN_23931557773761
// MI455X (gfx1250) — compile-verified
//
#include <hip/hip_runtime.h>
#include <math.h>

// Problem constants (match reference)
#define NN 50000
#define NE 800000
#define DD 64

typedef __attribute__((ext_vector_type(2))) float v2f;
typedef __attribute__((ext_vector_type(8))) float v8f;

// D = A(16x4,f32) * B(4x16,f32) + C(16x16,f32)  -- full fp32 WMMA, CDNA5
__device__ __forceinline__ v8f wmma4(v2f a, v2f b, v8f c) {
  return __builtin_amdgcn_wmma_f32_16x16x4_f32(
      /*neg_a=*/false, a, /*neg_b=*/false, b,
      /*c_mod=*/(short)0, c, /*reuse_a=*/false, /*reuse_b=*/false);
}

// LDS byte offset of a __shared__ object (generic -> addrspace(3) -> int).
__device__ __forceinline__ unsigned lds_off(const void* p) {
  return (unsigned)(unsigned long long)(const __attribute__((address_space(3))) void*)p;
}

// Stage a 64x64 f32 weight matrix (16 KB) into LDS with CDNA5 async copies:
// 128 threads x 8 chunks of B128. INST_OFFSET is added to BOTH the LDS dest
// and the global source address (CDNA5 ISA), so one base pair
// + immediate offsets covers all chunks. Tracked by ASYNCcnt.
__device__ __forceinline__ void async_copy_16k(const float* __restrict__ gsrc,
                                               float* __restrict__ ldst, int tid) {
  unsigned laddr = lds_off(ldst) + tid * 16;
  const float* g = gsrc + tid * 4;
  asm volatile("global_load_async_to_lds_b128 %0, %1, off offset:0"      :: "v"(laddr), "v"(g) : "memory");
  asm volatile("global_load_async_to_lds_b128 %0, %1, off offset:2048"   :: "v"(laddr), "v"(g) : "memory");
  asm volatile("global_load_async_to_lds_b128 %0, %1, off offset:4096"   :: "v"(laddr), "v"(g) : "memory");
  asm volatile("global_load_async_to_lds_b128 %0, %1, off offset:6144"   :: "v"(laddr), "v"(g) : "memory");
  asm volatile("global_load_async_to_lds_b128 %0, %1, off offset:8192"   :: "v"(laddr), "v"(g) : "memory");
  asm volatile("global_load_async_to_lds_b128 %0, %1, off offset:10240"  :: "v"(laddr), "v"(g) : "memory");
  asm volatile("global_load_async_to_lds_b128 %0, %1, off offset:12288"  :: "v"(laddr), "v"(g) : "memory");
  asm volatile("global_load_async_to_lds_b128 %0, %1, off offset:14336"  :: "v"(laddr), "v"(g) : "memory");
}

__device__ __forceinline__ void wait_async_all() {
  asm volatile("s_wait_asynccnt 0x0" ::: "memory");
}

__global__ void k_zero(float* __restrict__ p, int n) {
  int i = blockIdx.x * blockDim.x + threadIdx.x;
  if (i < n) p[i] = 0.0f;
}

__global__ void k_degree(const int* __restrict__ dst, float* __restrict__ cnt) {
  int e = blockIdx.x * blockDim.x + threadIdx.x;
  if (e < NE) atomicAdd(&cnt[dst[e]], 1.0f);
}

// in-place: cnt -> 1 / max(cnt, 1)
__global__ void k_invert(float* __restrict__ cnt) {
  int i = blockIdx.x * blockDim.x + threadIdx.x;
  if (i < NN) cnt[i] = 1.0f / fmaxf(cnt[i], 1.0f);
}

// One thread handles (edge, 4-float chunk): float4 gather (L2 resident)
// + 4 global f32 atomic adds into agg[dst].
__global__ void k_scatter(const float* __restrict__ h, int ldh,
                          const int* __restrict__ src, const int* __restrict__ dst,
                          float* __restrict__ agg) {
  int t = blockIdx.x * blockDim.x + threadIdx.x;
  int e = t >> 4;
  int c = t & 15;
  if (e >= NE) return;
  int s = src[e];
  int d = dst[e];
  const float4 v = *(const float4*)(h + (size_t)s * ldh + c * 4);
  float* a = agg + (size_t)d * DD + c * 4;
  atomicAdd(a + 0, v.x);
  atomicAdd(a + 1, v.y);
  atomicAdd(a + 2, v.z);
  atomicAdd(a + 3, v.w);
}

// Fused SAGE layer GEMM:  out = [agg*inv] @ Wl + h @ Wr + bias  (+optional ReLU)
// Grid: 3125 blocks x 128 threads = 4 waves/block; wave w handles output tile
// (tileM = blockIdx.x, tileN = w). Weights staged once per block into LDS via
// async global->LDS B128 copies; B fragments then come from ds_load.
// EXEC stays all-ones around every WMMA (no divergence).
__global__ void __launch_bounds__(128) k_sage_gemm(
    const float* __restrict__ agg, const float* __restrict__ inv,
    const float* __restrict__ h, int ldh,
    const float* __restrict__ Wl, const float* __restrict__ Wr,
    const float* __restrict__ bias,
    float* __restrict__ out, int ldo, int relu) {
  __shared__ float sWl[DD * DD];
  __shared__ float sWr[DD * DD];

  const int tid   = threadIdx.x;
  const int lane  = tid & 31;
  const int tileN = tid >> 5;           // 0..3  (64 cols / 16)
  const int tileM = blockIdx.x;         // 0..3124 (50000 rows / 16)
  const int l16   = lane & 15;
  const int half  = lane >> 4;
  const int rA    = tileM * 16 + l16;   // A-operand row for this lane
  const int nB    = tileN * 16 + l16;   // B/D column for this lane
  const int kb2   = 2 * half;           // K sub-offset per ISA fp32 layout

  // Stage weights into LDS (uniform branches; all lanes active).
  if (Wl != nullptr) async_copy_16k(Wl, sWl, tid);
  async_copy_16k(Wr, sWr, tid);
  wait_async_all();
  __syncthreads();

  v8f acc;
#pragma unroll
  for (int i = 0; i < 8; ++i) acc[i] = 0.0f;

  if (agg != nullptr) {                 // mean-aggregate operand (uniform branch)
    const float im = inv[rA];
#pragma unroll
    for (int k0 = 0; k0 < DD; k0 += 4) {
      const int k = k0 + kb2;
      v2f a, b;
      a.x = agg[(size_t)rA * DD + k] * im;
      a.y = agg[(size_t)rA * DD + k + 1] * im;
      b.x = sWl[k * DD + nB];
      b.y = sWl[(k + 1) * DD + nB];
      acc = wmma4(a, b, acc);
    }
  }
#pragma unroll
  for (int k0 = 0; k0 < DD; k0 += 4) {  // self (root) operand
    const int k = k0 + kb2;
    v2f a, b;
    a.x = h[(size_t)rA * ldh + k];
    a.y = h[(size_t)rA * ldh + k + 1];
    b.x = sWr[k * DD + nB];
    b.y = sWr[(k + 1) * DD + nB];
    acc = wmma4(a, b, acc);
  }

  const float bv = bias[nB];
#pragma unroll
  for (int r = 0; r < 8; ++r) {         // D layout: VGPR r -> M = r + 8*half
    float v = acc[r] + bv;
    if (relu) v = fmaxf(v, 0.0f);
    out[(size_t)(tileM * 16 + half * 8 + r) * ldo + nB] = v;
  }
}

// MLP head: hidden = relu(xi[N,256] @ W1[256,64] + b1); out = sigmoid(hidden @ W2 + b2)
// One wave (block of 32) per 16-row tile; 4 WMMA accumulator tiles across N=64,
// 64 K-steps of 16x16x4 => 256 WMMA per wave. Epilogue via LDS (padded stride 65).
__global__ void __launch_bounds__(32) k_mlp(
    const float* __restrict__ xi, const float* __restrict__ W1,
    const float* __restrict__ b1, const float* __restrict__ W2,
    const float* __restrict__ b2, float* __restrict__ out) {
  __shared__ float hid[16][65];
  const int lane  = threadIdx.x;        // 0..31
  const int tileM = blockIdx.x;         // 0..3124
  const int l16   = lane & 15;
  const int half  = lane >> 4;
  const int rA    = tileM * 16 + l16;
  const int kb2   = 2 * half;

  v8f acc[4];
#pragma unroll
  for (int t = 0; t < 4; ++t)
#pragma unroll
    for (int i = 0; i < 8; ++i) acc[t][i] = 0.0f;

  for (int k0 = 0; k0 < 4 * DD; k0 += 4) {
    const int k = k0 + kb2;
    v2f a;
    a.x = xi[(size_t)rA * 256 + k];
    a.y = xi[(size_t)rA * 256 + k + 1];
#pragma unroll
    for (int t = 0; t < 4; ++t) {
      v2f b;
      b.x = W1[(size_t)k * DD + t * 16 + l16];
      b.y = W1[(size_t)(k + 1) * DD + t * 16 + l16];
      acc[t] = wmma4(a, b, acc[t]);
    }
  }

#pragma unroll
  for (int t = 0; t < 4; ++t) {
    const float bv = b1[t * 16 + l16];
#pragma unroll
    for (int r = 0; r < 8; ++r) {
      hid[half * 8 + r][t * 16 + l16] = fmaxf(acc[t][r] + bv, 0.0f);
    }
  }
  __syncthreads();  // single-wave WG: DS in-order; barrier degrades to nop

  if (lane < 16) {
    float s = b2[0];
#pragma unroll 8
    for (int k = 0; k < DD; ++k) s += hid[lane][k] * W2[k];
    out[tileM * 16 + lane] = 1.0f / (1.0f + __expf(-s));
  }
}

extern "C" void kernel_launch(void* const* d_in, const int* in_sizes, int n_in,
                              void* d_out, int out_size, void* d_ws, size_t ws_size,
                              hipStream_t stream) {
  const float* x     = (const float*)d_in[0];
  const int*   eidx  = (const int*)d_in[1];   // [2, NE], row0 = src, row1 = dst
  const float* Wl    = (const float*)d_in[2]; // [3,64,64]
  const float* Wr    = (const float*)d_in[3]; // [3,64,64]
  const float* bl    = (const float*)d_in[4]; // [3,64]
  const float* Wskip = (const float*)d_in[5]; // [64,64]
  const float* bskip = (const float*)d_in[6]; // [64]
  const float* W1    = (const float*)d_in[7]; // [256,64]
  const float* b1    = (const float*)d_in[8]; // [64]
  const float* W2    = (const float*)d_in[9]; // [64,1]
  const float* b2    = (const float*)d_in[10];// [1]
  float* out = (float*)d_out;                 // [NN,1]

  const int* src = eidx;
  const int* dst = eidx + NE;

  // Workspace layout (~64.2 MB): xi concat buffer, aggregation buffer, degrees.
  float* xi  = (float*)d_ws;              // [NN, 256] : 4 blocks of 64 columns
  float* agg = xi + (size_t)NN * 256;     // [NN, 64]
  float* cnt = agg + (size_t)NN * DD;     // [NN] -> becomes 1/max(deg,1)

  // Degrees (computed once; graph fixed across layers).
  k_zero<<<(NN + 255) / 256, 256, 0, stream>>>(cnt, NN);
  k_degree<<<(NE + 255) / 256, 256, 0, stream>>>(dst, cnt);
  k_invert<<<(NN + 255) / 256, 256, 0, stream>>>(cnt);

  // xi block 0 = relu(x @ Wskip + bskip)  (single-operand path: agg == nullptr)
  k_sage_gemm<<<NN / 16, 128, 0, stream>>>(nullptr, nullptr, x, DD, nullptr,
                                           Wskip, bskip, xi + 0, 256, 1);

  const float* hcur = x;
  int ldh = DD;
  for (int layer = 0; layer < 3; ++layer) {
    k_zero<<<((NN * DD) + 255) / 256, 256, 0, stream>>>(agg, NN * DD);
    k_scatter<<<(NE * 16) / 256, 256, 0, stream>>>(hcur, ldh, src, dst, agg);
    float* ho  = xi + (size_t)(layer + 1) * DD;   // column block in concat buffer
    int relu   = (layer < 2) ? 1 : 0;
    k_sage_gemm<<<NN / 16, 128, 0, stream>>>(
        agg, cnt, hcur, ldh,
        Wl + (size_t)layer * DD * DD, Wr + (size_t)layer * DD * DD,
        bl + (size_t)layer * DD, ho, 256, relu);
    hcur = ho;
    ldh  = 256;
  }

  k_mlp<<<NN / 16, 32, 0, stream>>>(xi, W1, b1, W2, b2, out);
}